// MambaBlock_50603304682201
// MI455X (gfx1250) — compile-verified
//
#include <hip/hip_runtime.h>
#include <math.h>

typedef __attribute__((ext_vector_type(2))) float v2f;
typedef __attribute__((ext_vector_type(8))) float v8f;
typedef __attribute__((ext_vector_type(4))) unsigned int v4u;
typedef __attribute__((ext_vector_type(4))) int v4i_;
typedef __attribute__((ext_vector_type(8))) int v8i_;

#define L_TOK 2304   // 48*48 tokens
#define DM    32     // d_model
#define DI    2048   // d_inner
#define DX    4096   // 2*d_inner
#define NST   16     // d_state
#define NXDB  34     // dt_rank + 2*d_state
#define NDEPTH 4
#define KCHUNK 512   // K chunk staged by TDM into LDS for out_proj

__device__ __forceinline__ float silu_f(float x) { return x / (1.0f + __expf(-x)); }
__device__ __forceinline__ float gelu_f(float x) { return 0.5f * x * (1.0f + erff(x * 0.70710678118f)); }

// ---------------------------------------------------------------- project_in
__global__ void k_proj_in(const float* __restrict__ x, const float* __restrict__ w,
                          const float* __restrict__ b, const float* __restrict__ pos,
                          float* __restrict__ t) {
  int id = blockIdx.x * blockDim.x + threadIdx.x;
  if (id >= L_TOK * DM) return;
  int l = id >> 5, d = id & 31;
  float acc = b[d] + pos[d];
#pragma unroll
  for (int c = 0; c < 32; ++c) acc += x[c * L_TOK + l] * w[d * 32 + c];
  t[id] = acc;
}

// ------------------------------------------------- LayerNorm + in_proj (WMMA)
// grid (144, 64), block 128 (4 waves). Each wave -> one 16x16 tile of xz.
__global__ void __launch_bounds__(128)
k_ln_inproj(const float* __restrict__ t, const float* __restrict__ lnw,
            const float* __restrict__ lnb, const float* __restrict__ W,
            float* __restrict__ xz) {
  __shared__ float As[16][32];
  __shared__ float s_mu[16], s_rs[16];
  int tid = threadIdx.x;
  int l0 = blockIdx.x * 16;

  if (tid < 16) {
    const float* row = t + (size_t)(l0 + tid) * DM;
    float s = 0.f, ss = 0.f;
#pragma unroll
    for (int d = 0; d < 32; ++d) { float v = row[d]; s += v; ss += v * v; }
    float mu = s * (1.f / 32.f);
    float var = ss * (1.f / 32.f) - mu * mu;
    s_mu[tid] = mu;
    s_rs[tid] = rsqrtf(var + 1e-5f);
  }
  __syncthreads();
#pragma unroll
  for (int idx = tid; idx < 512; idx += 128) {
    int r = idx >> 5, d = idx & 31;
    As[r][d] = (t[(size_t)(l0 + r) * DM + d] - s_mu[r]) * s_rs[r] * lnw[d] + lnb[d];
  }
  __syncthreads();

  int lane = tid & 31, wv = tid >> 5;
  int n0 = blockIdx.y * 64 + wv * 16;
  int m  = lane & 15;
  int kh = (lane >> 4) * 2;                       // ISA 16x4 fp32 A layout
  const float* Wrow = W + (size_t)(n0 + m) * DM;  // B[k][n] = in_proj_w[n][k]
  v8f acc = {};
#pragma unroll
  for (int kb = 0; kb < 32; kb += 4) {
    v2f a, b;
    a.x = As[m][kb + kh];     a.y = As[m][kb + kh + 1];
    b.x = Wrow[kb + kh];      b.y = Wrow[kb + kh + 1];
    acc = __builtin_amdgcn_wmma_f32_16x16x4_f32(false, a, false, b, (short)0, acc, false, false);
  }
  int rbase = l0 + ((lane >> 4) << 3);            // C/D: vgpr r -> M = r + 8*hi
#pragma unroll
  for (int r = 0; r < 8; ++r)
    xz[(size_t)(rbase + r) * DX + n0 + m] = acc[r];
}

// --------------------------------------------- causal depthwise conv + SiLU
__global__ void k_conv_silu(const float* __restrict__ xz, const float* __restrict__ cw,
                            const float* __restrict__ cb, float* __restrict__ xc) {
  int id = blockIdx.x * blockDim.x + threadIdx.x;
  if (id >= L_TOK * DI) return;
  int l = id / DI, e = id % DI;
  float acc = cb[e];
#pragma unroll
  for (int j = 0; j < 4; ++j) {
    int ll = l - 3 + j;
    if (ll >= 0) acc += cw[e * 4 + j] * xz[(size_t)ll * DX + e];  // xi = cols [0,2048)
  }
  xc[id] = silu_f(acc);
}

// --------------------------------------------------- x_proj (wave reduction)
__global__ void k_xproj(const float* __restrict__ xc, const float* __restrict__ xpw,
                        float* __restrict__ xdb) {
  int wid  = (blockIdx.x * blockDim.x + threadIdx.x) >> 5;
  int lane = threadIdx.x & 31;
  if (wid >= L_TOK * NXDB) return;
  int l = wid / NXDB, r = wid % NXDB;
  const float* a = xc + (size_t)l * DI;
  const float* w = xpw + (size_t)r * DI;
  float s = 0.f;
  for (int k = lane; k < DI; k += 32) s += a[k] * w[k];
#pragma unroll
  for (int msk = 16; msk >= 1; msk >>= 1) s += __shfl_xor(s, msk, 32);
  if (lane == 0) xdb[wid] = s;
}

// -------------------------------------------------------- dt = softplus(...)
__global__ void k_dt(const float* __restrict__ xdb, const float* __restrict__ dtw,
                     const float* __restrict__ dtb, float* __restrict__ dt) {
  int id = blockIdx.x * blockDim.x + threadIdx.x;
  if (id >= L_TOK * DI) return;
  int l = id / DI, e = id % DI;
  float v = xdb[l * NXDB + 0] * dtw[e * 2 + 0] + xdb[l * NXDB + 1] * dtw[e * 2 + 1] + dtb[e];
  dt[id] = (v > 20.f) ? v : log1pf(__expf(v));
}

// -------------------------------------------- selective scan (seq over L)
// grid 128, block 256: thread = one (d,n) state element; shuffle-reduce over n.
__global__ void __launch_bounds__(256)
k_scan(const float* __restrict__ dt, const float* __restrict__ xc,
       const float* __restrict__ xdb, const float* __restrict__ A_log,
       float* __restrict__ y) {
  int tid = threadIdx.x;
  int n = tid & 15;
  int d = blockIdx.x * 16 + (tid >> 4);
  float A = -__expf(A_log[(size_t)d * NST + n]);
  float h = 0.f;
  for (int l = 0; l < L_TOK; ++l) {
    float dtv = dt[(size_t)l * DI + d];
    float xv  = xc[(size_t)l * DI + d];
    float bv  = xdb[l * NXDB + 2 + n];
    float cv  = xdb[l * NXDB + 2 + NST + n];
    h = h * __expf(dtv * A) + dtv * xv * bv;
    float yv = h * cv;
    yv += __shfl_xor(yv, 1, 32);
    yv += __shfl_xor(yv, 2, 32);
    yv += __shfl_xor(yv, 4, 32);
    yv += __shfl_xor(yv, 8, 32);
    if (n == 0) y[(size_t)l * DI + d] = yv;
  }
}

// ------------------------------------------------------------ gating (inplace)
__global__ void k_gate(float* __restrict__ y, const float* __restrict__ xc,
                       const float* __restrict__ xz, const float* __restrict__ Dp) {
  int id = blockIdx.x * blockDim.x + threadIdx.x;
  if (id >= L_TOK * DI) return;
  int l = id / DI, e = id % DI;
  float z = xz[(size_t)l * DX + DI + e];
  y[id] = (y[id] + xc[id] * Dp[e]) * silu_f(z);
}

// ---------- out_proj: TDM stages A (16 x KCHUNK fp32) into LDS, 8 waves split
// ---------- K, WMMA from LDS, cross-wave LDS reduce, fused GELU + residual.
// grid (144, 2), block 256.
struct OutSmem {
  float sA[16 * KCHUNK];   // 32 KB: TDM destination (offset 0 in group segment)
  float red[8][32][8];     // 8 KB: cross-wave reduction
};

__global__ void __launch_bounds__(256)
k_outproj(const float* __restrict__ yg, const float* __restrict__ opw,
          const float* __restrict__ gamma, float* __restrict__ t) {
  __shared__ OutSmem sm;
  int tid = threadIdx.x;
  int lane = tid & 31, wv = tid >> 5;
  int l0 = blockIdx.x * 16;
  int n0 = blockIdx.y * 16;
  int m = lane & 15, kh = (lane >> 4) * 2;
  const float* Brow = opw + (size_t)(n0 + m) * DI;   // B[k][n] = out_proj_w[n][k]
  v8f acc = {};

  for (int c = 0; c < DI / KCHUNK; ++c) {
    __syncthreads();                       // previous chunk fully consumed
    if (tid < 32) {                        // wave 0 drives the Tensor Data Mover
      unsigned long long ga =
          (unsigned long long)(uintptr_t)(yg + (size_t)l0 * DI + (size_t)c * KCHUNK);
      v4u g0;
      g0[0] = 1u;                                        // count=1, user descriptor
      g0[1] = 0u;                                        // lds_addr = &sm.sA (offset 0)
      g0[2] = (unsigned)(ga & 0xFFFFFFFFu);              // global_addr[31:0]
      g0[3] = (unsigned)((ga >> 32) & 0x01FFFFFFu)       // global_addr[56:32]
              | (2u << 30);                              // type=2 (image)
      v8i_ g1;
      g1[0] = (int)(2u << 16);               // workgroup_mask=0, data_size=2 (4B)
      g1[1] = (int)((unsigned)KCHUNK << 16); // tensor_dim0[15:0] @ bits 63:48
      g1[2] = (int)(16u << 16);              // tensor_dim1=16 @ bits 111:80
      g1[3] = (int)((unsigned)KCHUNK << 16); // tile_dim0 @ bits 127:112
      g1[4] = 16;                            // tile_dim1 @ bits 143:128
      g1[5] = DI;                            // tensor_dim0_stride=2048 @ bits 207:160
      g1[6] = 0;
      g1[7] = 0;
#if __clang_major__ >= 23
      __builtin_amdgcn_tensor_load_to_lds(g0, g1, (v4i_)0, (v4i_)0, (v8i_)0, 0);
#else
      __builtin_amdgcn_tensor_load_to_lds(g0, g1, (v4i_)0, (v4i_)0, 0);
#endif
      __builtin_amdgcn_s_wait_tensorcnt(0);
    }
    __syncthreads();                       // tile visible to all waves

    const float* Arow = &sm.sA[m * KCHUNK];      // TDM wrote row-major 16 x KCHUNK
    const float* Bc   = Brow + (size_t)c * KCHUNK;
    int kbase = wv * (KCHUNK / 8);               // 64-wide K slice per wave
#pragma unroll
    for (int it = 0; it < KCHUNK / 8 / 4; ++it) {   // 16 WMMA k-steps
      int kb = kbase + it * 4;
      __builtin_prefetch(&Bc[kb + 64], 0, 1);
      v2f a, b;
      a.x = Arow[kb + kh]; a.y = Arow[kb + kh + 1];
      b.x = Bc[kb + kh];   b.y = Bc[kb + kh + 1];
      acc = __builtin_amdgcn_wmma_f32_16x16x4_f32(false, a, false, b, (short)0, acc, false, false);
    }
  }

#pragma unroll
  for (int r = 0; r < 8; ++r) sm.red[wv][lane][r] = acc[r];
  __syncthreads();
  int ls = tid >> 3, r = tid & 7;
  float s = 0.f;
#pragma unroll
  for (int w = 0; w < 8; ++w) s += sm.red[w][ls][r];
  int row = l0 + r + ((ls >> 4) << 3);
  int col = n0 + (ls & 15);
  t[(size_t)row * DM + col] += gamma[col] * gelu_f(s);
}

// ---------------------------------------------------------------- project_out
__global__ void k_proj_out(const float* __restrict__ t, const float* __restrict__ w,
                           const float* __restrict__ b, float* __restrict__ out) {
  int id = blockIdx.x * blockDim.x + threadIdx.x;
  if (id >= 64 * L_TOK) return;
  int o = id / L_TOK, l = id % L_TOK;
  float acc = b[o];
#pragma unroll
  for (int d = 0; d < 32; ++d) acc += t[(size_t)l * DM + d] * w[o * 32 + d];
  out[id] = acc;
}

extern "C" void kernel_launch(void* const* d_in, const int* in_sizes, int n_in,
                              void* d_out, int out_size, void* d_ws, size_t ws_size,
                              hipStream_t stream) {
  const float* x    = (const float*)d_in[0];
  const float* piw  = (const float*)d_in[1];
  const float* pib  = (const float*)d_in[2];
  const float* pos  = (const float*)d_in[3];
  const float* lnw  = (const float*)d_in[4];
  const float* lnb  = (const float*)d_in[5];
  const float* ipw  = (const float*)d_in[6];
  const float* cw   = (const float*)d_in[7];
  const float* cb   = (const float*)d_in[8];
  const float* xpw  = (const float*)d_in[9];
  const float* dtw  = (const float*)d_in[10];
  const float* dtb  = (const float*)d_in[11];
  const float* Alog = (const float*)d_in[12];
  const float* Dp   = (const float*)d_in[13];
  const float* opw  = (const float*)d_in[14];
  const float* gma  = (const float*)d_in[15];
  const float* pow_ = (const float*)d_in[16];
  const float* pob  = (const float*)d_in[17];
  float* out = (float*)d_out;

  float* ws = (float*)d_ws;
  size_t o = 0;
  float* t_buf = ws + o; o += (size_t)L_TOK * DM;    // 73728
  float* xz    = ws + o; o += (size_t)L_TOK * DX;    // 9.44M
  float* xc    = ws + o; o += (size_t)L_TOK * DI;    // 4.72M
  float* xdb   = ws + o; o += (size_t)L_TOK * NXDB;  // 78336
  float* dt    = ws + o; o += (size_t)L_TOK * DI;    // 4.72M
  float* y     = ws + o;                             // 4.72M  (~91 MB total)

  k_proj_in<<<(L_TOK * DM + 255) / 256, 256, 0, stream>>>(x, piw, pib, pos, t_buf);

  for (int i = 0; i < NDEPTH; ++i) {
    k_ln_inproj<<<dim3(L_TOK / 16, DX / 64), 128, 0, stream>>>(
        t_buf, lnw + i * DM, lnb + i * DM, ipw + (size_t)i * DX * DM, xz);
    k_conv_silu<<<(L_TOK * DI + 255) / 256, 256, 0, stream>>>(
        xz, cw + (size_t)i * DI * 4, cb + (size_t)i * DI, xc);
    k_xproj<<<(L_TOK * NXDB + 7) / 8, 256, 0, stream>>>(
        xc, xpw + (size_t)i * NXDB * DI, xdb);
    k_dt<<<(L_TOK * DI + 255) / 256, 256, 0, stream>>>(
        xdb, dtw + (size_t)i * DI * 2, dtb + (size_t)i * DI, dt);
    k_scan<<<DI / 16, 256, 0, stream>>>(
        dt, xc, xdb, Alog + (size_t)i * DI * NST, y);
    k_gate<<<(L_TOK * DI + 255) / 256, 256, 0, stream>>>(
        y, xc, xz, Dp + (size_t)i * DI);
    k_outproj<<<dim3(L_TOK / 16, DM / 16), 256, 0, stream>>>(
        y, opw + (size_t)i * DM * DI, gma + i * DM, t_buf);
  }
  k_proj_out<<<(64 * L_TOK + 255) / 256, 256, 0, stream>>>(t_buf, pow_, pob, out);
}